// MRFCorrection_68238440398870
// MI455X (gfx1250) — compile-verified
//
#include <hip/hip_runtime.h>

#define H 128
#define LDSP 132  // LDS row pitch in floats (128 + 4 pad: kills 64-bank conflicts)

typedef float v2f __attribute__((ext_vector_type(2)));
typedef float v8f __attribute__((ext_vector_type(8)));

// ---------------------------------------------------------------------------
// Utility kernels
// ---------------------------------------------------------------------------
__global__ void mrf_fill_zero(float* __restrict__ p, long long n) {
    long long i = (long long)blockIdx.x * blockDim.x + threadIdx.x;
    long long stride = (long long)gridDim.x * blockDim.x;
    for (; i < n; i += stride) p[i] = 0.0f;
}

__global__ void mrf_copy(const float* __restrict__ s, float* __restrict__ d, long long n) {
    long long i = (long long)blockIdx.x * blockDim.x + threadIdx.x;
    long long stride = (long long)gridDim.x * blockDim.x;
    for (; i < n; i += stride) d[i] = s[i];
}

// ---------------------------------------------------------------------------
// FP32 WMMA GEMM:  Y[N,128] = X[N,128] @ W[128,128]
// One block = 256 threads = 8 waves; block covers 128 rows; each wave: 16 rows
// x full 128 cols (8 accumulator tiles). W staged transposed in LDS so both A
// and B fragments are contiguous float2 -> ds_load_b64.
//
// V_WMMA_F32_16X16X4_F32 fragment layout (ISA 7.12.2):
//   A 16x4:  lanes 0-15  -> M=lane,  v0=K0, v1=K1
//            lanes 16-31 -> M=lane-16, v0=K2, v1=K3
//   B 4x16:  v0: lanes0-15 K0,N=lane | lanes16-31 K2,N=lane-16
//            v1: lanes0-15 K1        | lanes16-31 K3
//   C/D 16x16: VGPR r: lanes0-15 M=r, lanes16-31 M=r+8, N=lane%16
// ---------------------------------------------------------------------------
__global__ __launch_bounds__(256)
void mrf_gemm_wmma_f32(const float* __restrict__ X, const float* __restrict__ W,
                       float* __restrict__ Y, int N) {
    __shared__ float lds_x[128 * LDSP];   // X tile, row-major, pitch 132
    __shared__ float lds_wt[H * LDSP];    // W transposed: lds_wt[c*LDSP + k] = W[k][c]

    const int tid = threadIdx.x;
    const int blockRow = blockIdx.x * 128;

    // Stage X tile (row index clamped; OOB rows masked at store time)
    for (int i = 0; i < 16; ++i) {
        int idx = tid + i * 256;           // 0..4095 float4 slots
        int row = idx >> 5;                // 0..127
        int c4  = (idx & 31) << 2;         // 0,4,...,124
        int gr = blockRow + row;
        if (gr >= N) gr = N - 1;
        float4 v = *(const float4*)(X + (long long)gr * H + c4);
        *(float4*)(&lds_x[row * LDSP + c4]) = v;
    }
    // Stage W transposed
    for (int i = 0; i < 16; ++i) {
        int idx = tid + i * 256;
        int k  = idx >> 5;
        int c4 = (idx & 31) << 2;
        float4 v = *(const float4*)(W + k * H + c4);
        lds_wt[(c4 + 0) * LDSP + k] = v.x;
        lds_wt[(c4 + 1) * LDSP + k] = v.y;
        lds_wt[(c4 + 2) * LDSP + k] = v.z;
        lds_wt[(c4 + 3) * LDSP + k] = v.w;
    }
    __syncthreads();

    const int wave = tid >> 5;
    const int lane = tid & 31;
    const int lo = lane & 15;
    const int hi = lane >> 4;             // 0 or 1
    const int r0 = wave * 16;

    v8f acc[8] = {};

    const float* xrow = &lds_x[(r0 + lo) * LDSP];
    for (int kb = 0; kb < H; kb += 4) {
        // A fragment: float2 at K = kb + 2*hi (8B aligned in LDS)
        v2f a = *(const v2f*)(xrow + kb + 2 * hi);
#pragma unroll
        for (int t = 0; t < 8; ++t) {
            const float* wcol = &lds_wt[(t * 16 + lo) * LDSP];
            v2f b = *(const v2f*)(wcol + kb + 2 * hi);
            acc[t] = __builtin_amdgcn_wmma_f32_16x16x4_f32(
                false, a, false, b, (short)0, acc[t], false, false);
        }
    }

    // Store D: acc[t][r] -> Y[blockRow + r0 + r + 8*hi][t*16 + lo]
#pragma unroll
    for (int t = 0; t < 8; ++t) {
        int col = t * 16 + lo;
#pragma unroll
        for (int r = 0; r < 8; ++r) {
            int grow = blockRow + r0 + r + 8 * hi;
            if (grow < N) Y[(long long)grow * H + col] = acc[t][r];
        }
    }
}

// ---------------------------------------------------------------------------
// Edge scatter: one wave per edge; lane handles 4 consecutive features.
// msgs[dst] += weighted[src]; cnt[dst] += 1
// ---------------------------------------------------------------------------
__global__ __launch_bounds__(256)
void mrf_scatter(const float* __restrict__ weighted,
                 const int* __restrict__ e_src, const int* __restrict__ e_dst,
                 float* __restrict__ msgs, float* __restrict__ cnt, int E) {
    int e = blockIdx.x * 8 + (threadIdx.x >> 5);
    if (e >= E) return;
    int lane = threadIdx.x & 31;
    int s = e_src[e];
    int d = e_dst[e];
    float4 v = *(const float4*)(weighted + (long long)s * H + lane * 4);
    float* mp = msgs + (long long)d * H + lane * 4;
    atomicAdd(mp + 0, v.x);
    atomicAdd(mp + 1, v.y);
    atomicAdd(mp + 2, v.z);
    atomicAdd(mp + 3, v.w);
    if (lane == 0) atomicAdd(cnt + d, 1.0f);
}

// ---------------------------------------------------------------------------
// cur += relu(where(cnt>0, msgs/cnt, 0) + bias)
// ---------------------------------------------------------------------------
__global__ __launch_bounds__(256)
void mrf_update(float* __restrict__ cur, const float* __restrict__ msgs,
                const float* __restrict__ cnt, const float* __restrict__ bias, int Nn) {
    long long idx = (long long)blockIdx.x * blockDim.x + threadIdx.x;
    long long total = (long long)Nn * (H / 4);
    if (idx >= total) return;
    long long node = idx >> 5;
    int c4 = (int)(idx & 31) << 2;
    float c = cnt[node];
    float inv = (c > 0.0f) ? (1.0f / c) : 0.0f;
    float4 m = *(const float4*)(msgs + node * H + c4);
    float4 bv = *(const float4*)(bias + c4);
    float4 o = *(float4*)(cur + node * H + c4);
    o.x += fmaxf(m.x * inv + bv.x, 0.0f);
    o.y += fmaxf(m.y * inv + bv.y, 0.0f);
    o.z += fmaxf(m.z * inv + bv.z, 0.0f);
    o.w += fmaxf(m.w * inv + bv.w, 0.0f);
    *(float4*)(cur + node * H + c4) = o;
}

// ---------------------------------------------------------------------------
extern "C" void kernel_launch(void* const* d_in, const int* in_sizes, int n_in,
                              void* d_out, int out_size, void* d_ws, size_t ws_size,
                              hipStream_t stream) {
    (void)n_in; (void)out_size; (void)ws_size;
    const float* x_a  = (const float*)d_in[0];
    const float* x_b  = (const float*)d_in[1];
    const int*   ei_ab = (const int*)d_in[2];
    const int*   ei_ba = (const int*)d_in[3];
    const float* W_ab = (const float*)d_in[4];
    const float* W_ba = (const float*)d_in[5];
    const float* b_a  = (const float*)d_in[6];
    const float* b_b  = (const float*)d_in[7];

    const int NA   = in_sizes[0] / H;
    const int NB   = in_sizes[1] / H;
    const int E_ab = in_sizes[2] / 2;
    const int E_ba = in_sizes[3] / 2;

    const long long na = (long long)NA * H;
    const long long nb = (long long)NB * H;

    float* out_a = (float*)d_out;          // cur_a lives in d_out
    float* out_b = out_a + na;             // cur_b

    float* ws    = (float*)d_ws;
    float* wgt_a = ws;                     // cur_a @ W_ab
    float* wgt_b = wgt_a + na;             // cur_b @ W_ba
    float* msg_a = wgt_b + nb;             // contiguous zero region starts here
    float* msg_b = msg_a + na;
    float* cnt_a = msg_b + nb;
    float* cnt_b = cnt_a + NA;

    // cur = x
    mrf_copy<<<2048, 256, 0, stream>>>(x_a, out_a, na);
    mrf_copy<<<2048, 256, 0, stream>>>(x_b, out_b, nb);

    for (int it = 0; it < 3; ++it) {
        // Jacobi: both GEMMs read pre-update cur_a / cur_b
        mrf_gemm_wmma_f32<<<(NA + 127) / 128, 256, 0, stream>>>(out_a, W_ab, wgt_a, NA);
        mrf_gemm_wmma_f32<<<(NB + 127) / 128, 256, 0, stream>>>(out_b, W_ba, wgt_b, NB);

        mrf_fill_zero<<<4096, 256, 0, stream>>>(msg_a, na + nb + NA + NB);

        // b -> a : weighted_b gathered by ei_ba[0], scattered to msg_a at ei_ba[1]
        mrf_scatter<<<(E_ba + 7) / 8, 256, 0, stream>>>(wgt_b, ei_ba, ei_ba + E_ba,
                                                        msg_a, cnt_a, E_ba);
        // a -> b
        mrf_scatter<<<(E_ab + 7) / 8, 256, 0, stream>>>(wgt_a, ei_ab, ei_ab + E_ab,
                                                        msg_b, cnt_b, E_ab);

        mrf_update<<<(int)((na / 4 + 255) / 256), 256, 0, stream>>>(out_a, msg_a, cnt_a, b_a, NA);
        mrf_update<<<(int)((nb / 4 + 255) / 256), 256, 0, stream>>>(out_b, msg_b, cnt_b, b_b, NB);
    }
}